// TestModel_88450556493885
// MI455X (gfx1250) — compile-verified
//
#include <hip/hip_runtime.h>
#include <hip/hip_bf16.h>
#include <math.h>

#define SPH_PI 3.14159265358979323846f

typedef float v2f __attribute__((ext_vector_type(2)));
typedef float v8f __attribute__((ext_vector_type(8)));

#define KL 32        // kh*kw = 4*8
#define NT 64        // pixels per block tile (4 waves x 16)
#define MT 64        // output channels per block tile (2 waves x 2 frags x 16)
#define SB_STRIDE 65 // padded LDS row stride to avoid bank conflicts

// ---------------------------------------------------------------------------
// Precompute great-circle sample coords (row,col) for [kh=4, kw=8, H, W].
// ---------------------------------------------------------------------------
__global__ void sph_coords_kernel(float2* __restrict__ coords, int H, int W, float delta_r) {
    int idx = blockIdx.x * blockDim.x + threadIdx.x;
    int HW = H * W;
    if (idx >= KL * HW) return;
    int kl  = idx / HW;
    int pix = idx - kl * HW;
    int k = kl >> 3;   // ring index 0..3
    int l = kl & 7;    // bearing index 0..7
    int h = pix / W;
    int w = pix - h * W;

    float lat = (0.5f - (h + 0.5f) / (float)H) * SPH_PI;
    float lon = ((w + 0.5f) / (float)W - 0.5f) * 2.0f * SPH_PI;
    float r   = delta_r * (float)(k + 1) / 4.0f;
    float th  = 2.0f * SPH_PI * (float)l / 8.0f;

    float sl = sinf(lat), cl = cosf(lat);
    float sr = sinf(r),   cr = cosf(r);
    float st = sinf(th),  ct = cosf(th);

    float sin_lat2 = sl * cr + cl * sr * ct;
    sin_lat2 = fminf(1.0f, fmaxf(-1.0f, sin_lat2));
    float lat2 = asinf(sin_lat2);
    float lon2 = lon + atan2f(st * sr * cl, cr - sl * sin_lat2);

    float rowv = (0.5f - lat2 / SPH_PI) * (float)H - 0.5f;
    float colv = (lon2 / (2.0f * SPH_PI) + 0.5f) * (float)W - 0.5f;
    coords[idx] = make_float2(rowv, colv);
}

// ---------------------------------------------------------------------------
// Fused bilinear-gather (im2col) + GEMM via V_WMMA_F32_16X16X4_F32.
// out[b,o,h,w] = sum_{c,kl} bilerp(x[b,c], coords[kl,h,w]) * wgt[o, c*32+kl] + bias[o]
// Block: 256 threads = 8 waves; tile M=64 (2 waves x 2 frags) x N=64 (4 waves).
// Double-buffered LDS B tile: gather of channel c+1 overlaps WMMA of channel c.
// grid.x = B*H*W / 64, grid.y = O / 64.
// ---------------------------------------------------------------------------
__global__ __launch_bounds__(256) void sph_conv_wmma_kernel(
    const float* __restrict__ x, const float* __restrict__ wgt,
    const float* __restrict__ bias, const float2* __restrict__ coords,
    float* __restrict__ out, int C, int O, int H, int W) {

    const int HW     = H * W;
    const int tid    = threadIdx.x;
    const int lane   = tid & 31;
    const int lane16 = lane & 15;
    const int klane  = lane >> 4;          // 0 or 1 (half-wave)
    const int wave   = tid >> 5;           // 0..7
    const int mWave  = wave >> 2;          // 2 waves along M
    const int nWave  = wave & 3;           // 4 waves along N

    const int n0      = blockIdx.x * NT;
    const int b       = n0 / HW;           // NT divides HW -> tile never crosses batch
    const int pixBase = n0 - b * HW;
    const int mTile   = blockIdx.y * MT + mWave * 32;   // this wave's 32 output rows
    const int nBase   = nWave * 16;                     // column within block tile
    const int K       = C * KL;

    // Bilinear metadata shared across channels (40 KB) + double B tile (16.6 KB)
    __shared__ int   sI00[KL * NT], sI10[KL * NT], sDC[KL * NT];
    __shared__ float sWr[KL * NT], sWc[KL * NT];
    __shared__ float sB[2][KL * SB_STRIDE];

    for (int e = tid; e < KL * NT; e += 256) {
        int kl  = e >> 6;      // NT == 64
        int col = e & 63;
        float2 rc = coords[kl * HW + pixBase + col];

        float r0f = floorf(rc.x);
        float wr  = rc.x - r0f;
        int r0 = (int)r0f;
        int r1 = r0 + 1;
        r0 = min(max(r0, 0), H - 1);
        r1 = min(max(r1, 0), H - 1);

        float c0f = floorf(rc.y);
        float wc  = rc.y - c0f;
        int c0 = (int)c0f % W; if (c0 < 0) c0 += W;   // wrap longitude
        int c1 = c0 + 1; if (c1 >= W) c1 -= W;

        sI00[e] = r0 * W + c0;
        sI10[e] = r1 * W + c0;
        sDC[e]  = c1 - c0;       // 1 or 1-W
        sWr[e]  = wr;
        sWc[e]  = wc;
    }

    // Gather B tile [32 kl x 64 pixels] for channel c into LDS buffer `buf`.
    auto gather = [&](int c, int buf) {
        const float* xc = x + (size_t)(b * C + c) * HW;
        for (int e = tid; e < KL * NT; e += 256) {
            float wr = sWr[e], wc = sWc[e];
            int i00 = sI00[e], i10 = sI10[e], dc = sDC[e];
            float v00 = xc[i00];
            float v01 = xc[i00 + dc];
            float v10 = xc[i10];
            float v11 = xc[i10 + dc];
            float v = (1.f - wr) * ((1.f - wc) * v00 + wc * v01)
                    +        wr  * ((1.f - wc) * v10 + wc * v11);
            int kl = e >> 6, col = e & 63;
            sB[buf][kl * SB_STRIDE + col] = v;
        }
    };

    v8f acc0 = {0.f, 0.f, 0.f, 0.f, 0.f, 0.f, 0.f, 0.f};
    v8f acc1 = {0.f, 0.f, 0.f, 0.f, 0.f, 0.f, 0.f, 0.f};
    const float* wrow0 = wgt + (size_t)(mTile + lane16) * K;        // frag0 rows
    const float* wrow1 = wrow0 + (size_t)16 * K;                    // frag1 rows (+16)

    __syncthreads();          // metadata visible
    gather(0, 0);
    __syncthreads();

    for (int c = 0; c < C; ++c) {
        int buf = c & 1;
        if (c + 1 < C) gather(c + 1, buf ^ 1);   // overlap next gather with WMMA

        const float* wr0 = wrow0 + c * KL;
        const float* wr1 = wrow1 + c * KL;
        #pragma unroll
        for (int ks = 0; ks < 8; ++ks) {
            int kk = ks * 4 + klane * 2;
            // A fragments (16x4 f32): lanes 0-15 carry K=k0,k0+1; lanes 16-31 K=k0+2,k0+3
            v2f a0 = *reinterpret_cast<const v2f*>(wr0 + kk);
            v2f a1 = *reinterpret_cast<const v2f*>(wr1 + kk);
            // B fragment (4x16 f32): v0 = row kk, v1 = row kk+1, column = lane16
            v2f bb;
            bb.x = sB[buf][kk * SB_STRIDE + nBase + lane16];
            bb.y = sB[buf][(kk + 1) * SB_STRIDE + nBase + lane16];
            acc0 = __builtin_amdgcn_wmma_f32_16x16x4_f32(
                false, a0, false, bb, (short)0, acc0, false, false);
            acc1 = __builtin_amdgcn_wmma_f32_16x16x4_f32(
                false, a1, false, bb, (short)0, acc1, false, false);
        }
        __syncthreads();
    }

    // C/D layout: VGPR j -> M = j (lanes 0-15) / j+8 (lanes 16-31), N = lane16
    #pragma unroll
    for (int j = 0; j < 8; ++j) {
        int o0 = mTile + j + klane * 8;
        int o1 = o0 + 16;
        size_t nIdx = (size_t)pixBase + nBase + lane16;
        out[(size_t)(b * O + o0) * HW + nIdx] = acc0[j] + bias[o0];
        out[(size_t)(b * O + o1) * HW + nIdx] = acc1[j] + bias[o1];
    }
}

// ---------------------------------------------------------------------------
// Per-channel mean/var over (B=2, H, W).  One block per channel.
// ---------------------------------------------------------------------------
__global__ void bn_stats_kernel(const float* __restrict__ x, float* __restrict__ mv,
                                int C, int HW) {
    int c = blockIdx.x;
    float s = 0.f, ss = 0.f;
    for (int b = 0; b < 2; ++b) {
        const float* p = x + (size_t)(b * C + c) * HW;
        for (int i = threadIdx.x; i < HW; i += blockDim.x) {
            float v = p[i];
            s += v; ss += v * v;
        }
    }
    __shared__ float rs[256], rss[256];
    rs[threadIdx.x] = s; rss[threadIdx.x] = ss;
    __syncthreads();
    for (int st = 128; st > 0; st >>= 1) {
        if (threadIdx.x < st) {
            rs[threadIdx.x]  += rs[threadIdx.x + st];
            rss[threadIdx.x] += rss[threadIdx.x + st];
        }
        __syncthreads();
    }
    if (threadIdx.x == 0) {
        float n = (float)(2 * HW);
        float mean = rs[0] / n;
        float var  = rss[0] / n - mean * mean;
        mv[c]     = mean;
        mv[C + c] = fmaxf(var, 0.f);
    }
}

// ---------------------------------------------------------------------------
// Fused BN -> ReLU -> 2x2 maxpool.  H,W are the pre-pool dims.
// ---------------------------------------------------------------------------
__global__ void bn_relu_pool_kernel(const float* __restrict__ x, const float* __restrict__ mv,
                                    const float* __restrict__ gamma, const float* __restrict__ beta,
                                    float* __restrict__ y, int C, int H, int W) {
    int Ho = H >> 1, Wo = W >> 1;
    int total = 2 * C * Ho * Wo;
    int idx = blockIdx.x * blockDim.x + threadIdx.x;
    if (idx >= total) return;
    int wo = idx % Wo; int t = idx / Wo;
    int ho = t % Ho;   t /= Ho;
    int c  = t % C;    int b = t / C;

    const float* p = x + (size_t)(b * C + c) * H * W;
    float mean = mv[c];
    float inv  = rsqrtf(mv[C + c] + 1e-5f);
    float g = gamma[c], be = beta[c];
    int r = 2 * ho, cc = 2 * wo;

    float v00 = fmaxf((p[r * W + cc]           - mean) * inv * g + be, 0.f);
    float v01 = fmaxf((p[r * W + cc + 1]       - mean) * inv * g + be, 0.f);
    float v10 = fmaxf((p[(r + 1) * W + cc]     - mean) * inv * g + be, 0.f);
    float v11 = fmaxf((p[(r + 1) * W + cc + 1] - mean) * inv * g + be, 0.f);
    y[idx] = fmaxf(fmaxf(v00, v01), fmaxf(v10, v11));
}

// ---------------------------------------------------------------------------
// BN -> ReLU (no pool), elementwise.
// ---------------------------------------------------------------------------
__global__ void bn_relu_kernel(const float* __restrict__ x, const float* __restrict__ mv,
                               const float* __restrict__ gamma, const float* __restrict__ beta,
                               float* __restrict__ y, int C, int HW) {
    int total = 2 * C * HW;
    int idx = blockIdx.x * blockDim.x + threadIdx.x;
    if (idx >= total) return;
    int c = (idx / HW) % C;
    float mean = mv[c];
    float inv  = rsqrtf(mv[C + c] + 1e-5f);
    y[idx] = fmaxf((x[idx] - mean) * inv * gamma[c] + beta[c], 0.f);
}

// ---------------------------------------------------------------------------
extern "C" void kernel_launch(void* const* d_in, const int* in_sizes, int n_in,
                              void* d_out, int out_size, void* d_ws, size_t ws_size,
                              hipStream_t stream) {
    const float* image = (const float*)d_in[0];
    const float* w1 = (const float*)d_in[1];  const float* b1 = (const float*)d_in[2];
    const float* g1 = (const float*)d_in[3];  const float* be1 = (const float*)d_in[4];
    const float* w2 = (const float*)d_in[5];  const float* b2 = (const float*)d_in[6];
    const float* g2 = (const float*)d_in[7];  const float* be2 = (const float*)d_in[8];
    const float* w3 = (const float*)d_in[9];  const float* b3 = (const float*)d_in[10];
    const float* g3 = (const float*)d_in[11]; const float* be3 = (const float*)d_in[12];
    const float* w4 = (const float*)d_in[13]; const float* b4 = (const float*)d_in[14];
    float* outp = (float*)d_out;

    const float D1 = (float)(M_PI / 150.0 * 7.0);
    const float D2 = (float)(M_PI / 75.0 * 5.0);
    const float D3 = (float)(M_PI / 37.0 * 5.0);   // D4 == D3, coords reused

    // Bump allocation inside workspace
    char* ws = (char*)d_ws;
    size_t off = 0;
    auto take = [&](size_t bytes) -> void* {
        void* p = ws + off;
        off = (off + bytes + 255) & ~(size_t)255;
        return p;
    };
    float2* coords1 = (float2*)take((size_t)KL * 256 * 512 * sizeof(float2)); // 33.6 MB
    float2* coords2 = (float2*)take((size_t)KL * 128 * 256 * sizeof(float2)); //  8.4 MB
    float2* coords3 = (float2*)take((size_t)KL *  64 * 128 * sizeof(float2)); //  2.1 MB
    float*  bufA    = (float*)take((size_t)2 * 64 * 256 * 512 * 4);           // 67 MB (raw1/raw2/raw3)
    float*  bufB    = (float*)take((size_t)2 * 64 * 128 * 256 * 4);           // 16.8 MB (pool1/act3)
    float*  bufC    = (float*)take((size_t)2 * 128 * 64 * 128 * 4);           //  8.4 MB (pool2)
    float*  mv      = (float*)take((size_t)2 * 128 * 4);                      // mean|var

    // Coords for each resolution
    sph_coords_kernel<<<(KL * 256 * 512 + 255) / 256, 256, 0, stream>>>(coords1, 256, 512, D1);
    sph_coords_kernel<<<(KL * 128 * 256 + 255) / 256, 256, 0, stream>>>(coords2, 128, 256, D2);
    sph_coords_kernel<<<(KL *  64 * 128 + 255) / 256, 256, 0, stream>>>(coords3,  64, 128, D3);

    // Layer 1: conv(3->64) @ 256x512, BN+ReLU+pool -> [2,64,128,256]
    sph_conv_wmma_kernel<<<dim3(2 * 256 * 512 / NT, 64 / MT), 256, 0, stream>>>(
        image, w1, b1, coords1, bufA, 3, 64, 256, 512);
    bn_stats_kernel<<<64, 256, 0, stream>>>(bufA, mv, 64, 256 * 512);
    bn_relu_pool_kernel<<<(2 * 64 * 128 * 256 + 255) / 256, 256, 0, stream>>>(
        bufA, mv, g1, be1, bufB, 64, 256, 512);

    // Layer 2: conv(64->128) @ 128x256, BN+ReLU+pool -> [2,128,64,128]
    sph_conv_wmma_kernel<<<dim3(2 * 128 * 256 / NT, 128 / MT), 256, 0, stream>>>(
        bufB, w2, b2, coords2, bufA, 64, 128, 128, 256);
    bn_stats_kernel<<<128, 256, 0, stream>>>(bufA, mv, 128, 128 * 256);
    bn_relu_pool_kernel<<<(2 * 128 * 64 * 128 + 255) / 256, 256, 0, stream>>>(
        bufA, mv, g2, be2, bufC, 128, 128, 256);

    // Layer 3: conv(128->128) @ 64x128, BN+ReLU
    sph_conv_wmma_kernel<<<dim3(2 * 64 * 128 / NT, 128 / MT), 256, 0, stream>>>(
        bufC, w3, b3, coords3, bufA, 128, 128, 64, 128);
    bn_stats_kernel<<<128, 256, 0, stream>>>(bufA, mv, 128, 64 * 128);
    bn_relu_kernel<<<(2 * 128 * 64 * 128 + 255) / 256, 256, 0, stream>>>(
        bufA, mv, g3, be3, bufB, 128, 64 * 128);

    // Layer 4: conv(128->128) @ 64x128 -> d_out (bias only, no BN)
    sph_conv_wmma_kernel<<<dim3(2 * 64 * 128 / NT, 128 / MT), 256, 0, stream>>>(
        bufB, w4, b4, coords3, outp, 128, 128, 64, 128);

    (void)in_sizes; (void)n_in; (void)out_size; (void)ws_size;
}